// BilinearInteractionV2_76106820485390
// MI455X (gfx1250) — compile-verified
//
#include <hip/hip_runtime.h>

typedef __attribute__((ext_vector_type(2))) float v2f;
typedef __attribute__((ext_vector_type(8))) float v8f;

// Problem constants (match reference)
constexpr int BATCH = 2048;
constexpr int NFIELDS = 40;
constexpr int DIM = 64;      // EMB_DIM
constexpr int PAIRS = 780;

// Tiling
constexpr int BM = 128;          // batch rows per workgroup
constexpr int WROWS = 16;        // batch rows per wave
constexpr int A_STRIDE = 68;     // padded LDS stride for L tile (conflict-free A-frag reads)
constexpr int WB_STRIDE = 160;   // floats per K-row-pair of interleaved W
                                 //  (160 % 64 == 32 -> hi/lo lane groups hit disjoint bank windows)

__global__ __launch_bounds__(256, 2)
void bilinear_wmma_f32_kernel(const float* __restrict__ emb,   // [BATCH, NFIELDS, DIM]
                              const float* __restrict__ Wmat,  // [PAIRS, DIM, DIM]
                              const int*   __restrict__ lidx,  // [PAIRS]
                              const int*   __restrict__ ridx,  // [PAIRS]
                              float* __restrict__ out)         // [BATCH, PAIRS, DIM]
{
    // W stored K-row-pair interleaved: element (K, n) -> sWb[(K>>1)*WB_STRIDE + 2*n + (K&1)]
    __shared__ float sWb[(DIM / 2) * WB_STRIDE]; // 32*160*4 = 20480 B
    __shared__ float sA[BM * A_STRIDE];          // 128*68*4 = 34816 B (total ~54 KB of 320 KB/WGP)

    const int p     = blockIdx.x;          // pair id
    const int bbase = blockIdx.y * BM;     // batch tile base
    const int lf    = lidx[p];
    const int rf    = ridx[p];

    const int tid  = threadIdx.x;
    const int wave = tid >> 5;
    const int lane = tid & 31;
    const int hi   = lane >> 4;            // lane group (0: lanes 0-15, 1: lanes 16-31)
    const int m    = lane & 15;            // row-in-tile (A) / col-in-tile (B,C,D)

    // ---- Stage W[p] (64x64) into LDS, row-pair interleaved.
    // i -> (rp = K/2 in [0,32), c = column in [0,64)); one ds_store_b64 per element pair.
    const float* Wp = Wmat + p * DIM * DIM;
    #pragma unroll
    for (int i = tid; i < (DIM / 2) * DIM; i += 256) {
        const int rp = i >> 6;             // K-row pair
        const int c  = i & 63;             // column
        v2f w;
        w.x = Wp[(2 * rp + 0) * DIM + c];  // coalesced across lanes (c consecutive)
        w.y = Wp[(2 * rp + 1) * DIM + c];
        *(v2f*)&sWb[rp * WB_STRIDE + 2 * c] = w;   // banks {2c, 2c+1}: conflict-free
    }

    // ---- Stage L tile (128 rows x 64) into LDS, coalesced float4 loads ----
    #pragma unroll
    for (int i = tid; i < BM * 16; i += 256) {
        const int row  = i >> 4;
        const int col4 = i & 15;
        const float4 v = *(const float4*)(emb + ((bbase + row) * NFIELDS + lf) * DIM + col4 * 4);
        *(float4*)&sA[row * A_STRIDE + col4 * 4] = v;
    }
    __syncthreads();

    // ---- A fragments: 16 K-steps of 16x4 f32.
    //   lane (hi,m): M = m; VGPR0 = K=4k+2*hi, VGPR1 = K=4k+2*hi+1  -> contiguous b64 in LDS
    const int arow = wave * WROWS + m;
    v2f a[16];
    #pragma unroll
    for (int k = 0; k < 16; ++k)
        a[k] = *(const v2f*)&sA[arow * A_STRIDE + 4 * k + 2 * hi];

    // Per-lane B base: fragment (k, ct) sits at immediate offset (k*320 + ct*32) floats
    const float* bbase_ptr = &sWb[hi * WB_STRIDE + 2 * m];

    // ---- 4 column tiles of 16 output columns each ----
    #pragma unroll
    for (int ct = 0; ct < 4; ++ct) {
        v8f c = {};
        #pragma unroll
        for (int k = 0; k < 16; ++k) {
            // B fragment (4x16): lane (hi,m): N = m; VGPR0 = row K=4k+2*hi, VGPR1 = K=4k+2*hi+1.
            // Interleaved layout -> single contiguous b64 load, no register repacking.
            const v2f b = *(const v2f*)&bbase_ptr[k * (2 * WB_STRIDE) + ct * 32];
            c = __builtin_amdgcn_wmma_f32_16x16x4_f32(
                    /*neg_a=*/false, a[k], /*neg_b=*/false, b,
                    /*c_mod=*/(short)0, c, /*reuse_a=*/false, /*reuse_b=*/false);
        }

        // ---- Fuse Hadamard with right embedding and store.
        // C/D layout: VGPR r holds M = r + 8*hi, N = m.
        const int col = ct * 16 + m;
        #pragma unroll
        for (int r = 0; r < 8; ++r) {
            const int row = bbase + wave * WROWS + 8 * hi + r;
            const float rv = emb[(row * NFIELDS + rf) * DIM + col];
            out[(row * PAIRS + p) * DIM + col] = c[r] * rv;
        }
    }
}

extern "C" void kernel_launch(void* const* d_in, const int* in_sizes, int n_in,
                              void* d_out, int out_size, void* d_ws, size_t ws_size,
                              hipStream_t stream) {
    const float* emb  = (const float*)d_in[0];   // feature_emb [2048,40,64] f32
    const float* Wmat = (const float*)d_in[1];   // bilinear_W  [780,64,64]  f32
    const int*   li   = (const int*)d_in[2];     // left_idx  [780] i32
    const int*   ri   = (const int*)d_in[3];     // right_idx [780] i32
    float* out = (float*)d_out;                  // [2048,780,64] f32

    dim3 grid(PAIRS, BATCH / BM);                // (780, 16)
    bilinear_wmma_f32_kernel<<<grid, 256, 0, stream>>>(emb, Wmat, li, ri, out);
}